// BEVFeatureExtractorV2_12558484374043
// MI455X (gfx1250) — compile-verified
//
#include <hip/hip_runtime.h>
#include <hip/hip_bf16.h>

// BEVFeatureExtractorV2: bilinear gather of 5 box points per ROI from a
// (B=4, C=256, H=256, W=256) f32 feature map. Gather-bound; no matmul
// structure -> CDNA5 path used here is async global->LDS gather (ASYNCcnt)
// pipelined across the 5 points.

#define BATCH 4
#define NROI  512
#define NCH   256
#define FH    256
#define FW    256
#define NPT   5

typedef __attribute__((address_space(3))) void lds_void_t;

// Byte offset of a __shared__ object within the workgroup's LDS segment.
__device__ __forceinline__ unsigned lds_byte_off(void* p) {
  return (unsigned)(unsigned long long)(lds_void_t*)p;
}

// CDNA5 async gather: per-lane global address -> per-lane LDS address.
// GV mode (64-bit vaddr, saddr = off). Tracked with ASYNCcnt.
__device__ __forceinline__ void async_load_b64_to_lds(unsigned lds_byte,
                                                      const float* gaddr) {
  asm volatile("global_load_async_to_lds_b64 %0, %1, off"
               :: "v"(lds_byte), "v"(gaddr)
               : "memory");
}

#define WAIT_ASYNCCNT_LE(n) asm volatile("s_wait_asynccnt " #n ::: "memory")

__global__ __launch_bounds__(NCH) void bev_feat_extract_kernel(
    const float* __restrict__ feats,   // (B, C, H, W)
    const float* __restrict__ rois,    // (B, N, 7)
    float* __restrict__ out)           // (B, N, 5*C)
{
  // [point][channel] = {f(y0,xl), f(y0,xl+1), f(y1,xl), f(y1,xl+1)}  (20 KB)
  __shared__ float4 stage[NPT * NCH];

  const int c = threadIdx.x;      // channel
  const int n = blockIdx.x;       // roi
  const int b = blockIdx.y;       // batch

  const float* roi = rois + ((size_t)b * NROI + n) * 7;
  const float cx = roi[0];
  const float cy = roi[1];
  const float dx = roi[3];
  const float dy = roi[4];
  const float ry = roi[6];

  const float cr = cosf(ry);
  const float sr = sinf(ry);

  // Reference: corners_norm rows (-.5,-.5), (-.5,.5), (.5,.5), (.5,-.5),
  // scaled by dims, row-vector times R = [[c,-s],[s,c]], plus center.
  const float hx = 0.5f * dx, hy = 0.5f * dy;
  const float lkx[4] = { -hx, -hx,  hx,  hx };
  const float lky[4] = { -hy,  hy,  hy, -hy };
  float corx[4], cory[4];
#pragma unroll
  for (int i = 0; i < 4; ++i) {
    corx[i] =  lkx[i] * cr + lky[i] * sr + cx;
    cory[i] = -lkx[i] * sr + lky[i] * cr + cy;
  }

  // Point order matches reference output: center, front, back, left, right.
  float ptx[NPT], pty[NPT];
  ptx[0] = cx;                      pty[0] = cy;
  ptx[1] = (corx[0] + corx[1]) * 0.5f;  pty[1] = (cory[0] + cory[1]) * 0.5f;
  ptx[2] = (corx[2] + corx[3]) * 0.5f;  pty[2] = (cory[2] + cory[3]) * 0.5f;
  ptx[3] = (corx[0] + corx[3]) * 0.5f;  pty[3] = (cory[0] + cory[3]) * 0.5f;
  ptx[4] = (corx[1] + corx[2]) * 0.5f;  pty[4] = (cory[1] + cory[2]) * 0.5f;

  bool  selx0[NPT], selx1[NPT];
  float wA[NPT], wB[NPT], wC[NPT], wD[NPT];

  const float* chbase = feats + (size_t)(b * NCH + c) * (FH * FW);
  const unsigned lds0 = lds_byte_off(&stage[0]);

#pragma unroll
  for (int p = 0; p < NPT; ++p) {
    const float xs = (ptx[p] - (-51.2f)) / 0.1f / 4.0f;
    const float ys = (pty[p] - (-51.2f)) / 0.1f / 4.0f;
    const int xf = (int)floorf(xs);
    const int yf = (int)floorf(ys);
    const int x0 = min(max(xf, 0),     FW - 1);
    const int x1 = min(max(xf + 1, 0), FW - 1);
    const int y0 = min(max(yf, 0),     FH - 1);
    const int y1 = min(max(yf + 1, 0), FH - 1);
    // Load an x-pair [xl, xl+1] with one b64; selects below reproduce the
    // clamped-corner semantics exactly (and keep the pair load in-bounds).
    const int xl = min(x0, FW - 2);

    const float x0f = (float)x0, x1f = (float)x1;
    const float y0f = (float)y0, y1f = (float)y1;
    wA[p] = (x1f - xs) * (y1f - ys);   // pairs with f(y0,x0)
    wB[p] = (x1f - xs) * (ys - y0f);   // pairs with f(y1,x0)
    wC[p] = (xs - x0f) * (y1f - ys);   // pairs with f(y0,x1)
    wD[p] = (xs - x0f) * (ys - y0f);   // pairs with f(y1,x1)
    selx0[p] = (x0 == xl);
    selx1[p] = (x1 == xl);

    const unsigned l = lds0 + (unsigned)((p * NCH + c) * 16);
    async_load_b64_to_lds(l,     chbase + (size_t)y0 * FW + xl);
    async_load_b64_to_lds(l + 8, chbase + (size_t)y1 * FW + xl);
  }

  float* outp = out + (size_t)(b * NROI + n) * (NPT * NCH);

  // Async loads complete in order: after issuing 10, point p is ready once
  // ASYNCcnt <= 10 - 2*(p+1). Consume pipelined against remaining gathers.
#define CONSUME(p)                                                         \
  {                                                                        \
    const float4 v = stage[(p) * NCH + c];                                 \
    const float f00 = selx0[p] ? v.x : v.y;  /* (y0,x0) */                 \
    const float f01 = selx1[p] ? v.x : v.y;  /* (y0,x1) */                 \
    const float f10 = selx0[p] ? v.z : v.w;  /* (y1,x0) */                 \
    const float f11 = selx1[p] ? v.z : v.w;  /* (y1,x1) */                 \
    outp[(p) * NCH + c] =                                                  \
        f00 * wA[p] + f10 * wB[p] + f01 * wC[p] + f11 * wD[p];             \
  }

  WAIT_ASYNCCNT_LE(8); CONSUME(0);
  WAIT_ASYNCCNT_LE(6); CONSUME(1);
  WAIT_ASYNCCNT_LE(4); CONSUME(2);
  WAIT_ASYNCCNT_LE(2); CONSUME(3);
  WAIT_ASYNCCNT_LE(0); CONSUME(4);
#undef CONSUME
}

extern "C" void kernel_launch(void* const* d_in, const int* in_sizes, int n_in,
                              void* d_out, int out_size, void* d_ws, size_t ws_size,
                              hipStream_t stream) {
  const float* feats = (const float*)d_in[0];   // (4, 256, 256, 256) f32
  const float* rois  = (const float*)d_in[1];   // (4, 512, 7) f32
  float* out = (float*)d_out;                   // (4, 512, 1280) f32
  (void)in_sizes; (void)n_in; (void)out_size; (void)d_ws; (void)ws_size;

  dim3 grid(NROI, BATCH);
  bev_feat_extract_kernel<<<grid, NCH, 0, stream>>>(feats, rois, out);
}